// DGM_d_77421080477833
// MI455X (gfx1250) — compile-verified
//
#include <hip/hip_runtime.h>

typedef __attribute__((ext_vector_type(2))) float v2f;
typedef __attribute__((ext_vector_type(4))) float v4f;
typedef __attribute__((ext_vector_type(8))) float v8f;

#define BATCH 8
#define NPTS  4096
#define DIM   256
#define TOPK  10

// ---------------------------------------------------------------------------
// Sorted-insert into per-lane ascending top-K list (register resident after
// full unroll; select-chain form, no dynamic indexing).
// ---------------------------------------------------------------------------
__device__ __forceinline__ void topk_insert(float* bv, int* bi, float v, int idx) {
    if (v >= bv[TOPK - 1]) return;
#pragma unroll
    for (int j = TOPK - 1; j >= 0; --j) {
        float pj = bv[j];
        int   ij = bi[j];
        bool lt     = v < pj;
        bool ltprev = (j > 0) ? (v < bv[j - 1]) : false;
        bv[j] = lt ? (ltprev ? bv[j - 1] : v)   : pj;
        bi[j] = lt ? (ltprev ? bi[j - 1] : idx) : ij;
    }
}

// ---------------------------------------------------------------------------
// Async Global->LDS staging helpers (CDNA5 GLOBAL_LOAD_ASYNC_TO_LDS_B128,
// GVS addressing: lds-addr VGPR, 32-bit byte-offset VGPR, 64-bit SGPR base).
// Tracked by ASYNCcnt; pair with s_wait_asynccnt before consuming.
// ---------------------------------------------------------------------------
__device__ __forceinline__ unsigned lds_offset_of(const void* p) {
    // addrspace(3) offsets occupy the low 32 bits of the generic address
    return (unsigned)(uintptr_t)p;
}

__device__ __forceinline__ void async_copy_b128(unsigned lds_addr, unsigned byte_off,
                                                const void* sbase) {
    asm volatile("global_load_async_to_lds_b128 %0, %1, %2"
                 :
                 : "v"(lds_addr), "v"(byte_off), "s"(sbase)
                 : "memory");
}

__device__ __forceinline__ void async_wait0() {
    asm volatile("s_wait_asynccnt 0" ::: "memory");
}

// stage `nbytes` (multiple of 64*16) from gbase into LDS at lds_base
__device__ __forceinline__ void async_stage(const float* gbase, unsigned lds_base,
                                            int tid, int nthreads, int nbytes) {
    for (int off = tid * 16; off < nbytes; off += nthreads * 16)
        async_copy_b128(lds_base + (unsigned)off, (unsigned)off, (const void*)gbase);
}

// ---------------------------------------------------------------------------
// Kernel A: y = x @ W per batch with V_WMMA_F32_16X16X4_F32, plus per-row
// squared norms. Each WG (128 thr = 4 waves) owns 16 rows x all 256 cols,
// so sq is complete per-WG (no global atomics).
// ---------------------------------------------------------------------------
__launch_bounds__(128)
__global__ void embed_gemm_kernel(const float* __restrict__ x,
                                  const float* __restrict__ W,
                                  float* __restrict__ y,
                                  float* __restrict__ sq) {
    __shared__ __attribute__((aligned(16))) float As[16][32];   // 2 KB  (rows x K-chunk)
    __shared__ __attribute__((aligned(16))) float Wt[256][32];  // 32 KB (col-major chunk of W)
    __shared__ float sqacc[16];

    const int tid     = threadIdx.x;       // 0..127
    const int wave    = tid >> 5;          // 0..3
    const int lane    = tid & 31;
    const int half    = lane >> 4;         // 0/1 selects K-pair (A) / row-half (C)
    const int l16     = lane & 15;
    const int batch   = blockIdx.y;
    const int rowbase = blockIdx.x * 16;

    const float* xrow = x + ((size_t)batch * NPTS + rowbase) * DIM;

    if (tid < 16) sqacc[tid] = 0.0f;

    const v8f vzero = {0.f, 0.f, 0.f, 0.f, 0.f, 0.f, 0.f, 0.f};
    v8f acc[4];
#pragma unroll
    for (int t = 0; t < 4; ++t) acc[t] = vzero;

    for (int kc = 0; kc < DIM / 32; ++kc) {
        // stage A chunk (16x32): one float4 per thread
        {
            int r  = tid >> 3;
            int k4 = (tid & 7) * 4;
            *(v4f*)&As[r][k4] = *(const v4f*)(xrow + r * DIM + kc * 32 + k4);
        }
        // stage W chunk transposed: Wt[col][k], 32x256 floats
        for (int i = 0; i < 16; ++i) {
            int li4 = tid + i * 128;          // float4 index over (k, c)
            int li  = li4 * 4;
            int k   = li >> 8;
            int c   = li & 255;
            v4f g = *(const v4f*)(W + (size_t)(kc * 32 + k) * DIM + c);
            Wt[c + 0][k] = g.x; Wt[c + 1][k] = g.y;
            Wt[c + 2][k] = g.z; Wt[c + 3][k] = g.w;
        }
        __syncthreads();
#pragma unroll
        for (int ks = 0; ks < 8; ++ks) {
            int kloc = ks * 4 + half * 2;     // K pair per ISA A-fragment layout
            v2f a = *(const v2f*)&As[l16][kloc];
#pragma unroll
            for (int t = 0; t < 4; ++t) {
                int col = (wave * 4 + t) * 16 + l16;
                v2f b = *(const v2f*)&Wt[col][kloc];
                acc[t] = __builtin_amdgcn_wmma_f32_16x16x4_f32(
                    false, a, false, b, (short)0, acc[t], false, false);
            }
        }
        __syncthreads();
    }

    // write y and accumulate row norms
#pragma unroll
    for (int v = 0; v < 8; ++v) {
        int rloc = half * 8 + v;
        int row  = rowbase + rloc;
        float ss = 0.0f;
#pragma unroll
        for (int t = 0; t < 4; ++t) {
            int col = (wave * 4 + t) * 16 + l16;
            float val = acc[t][v];
            y[((size_t)batch * NPTS + row) * DIM + col] = val;
            ss += val * val;
        }
        atomicAdd(&sqacc[rloc], ss);
    }
    __syncthreads();
    if (tid < 16) sq[batch * NPTS + rowbase + tid] = sqacc[tid];
}

// ---------------------------------------------------------------------------
// Kernel B: fused Gram + top-K.  WG = 64 thr (2 waves), 32 rows/WG.
// B tiles are double-buffered in LDS and streamed with async Global->LDS
// loads so the 16 KB fetch of tile jt+1 hides under tile jt's 64 WMMAs.
// Candidates ranked by (sq_j - 2G) (order-equivalent to lq); two lanes per
// row keep partial top-10 lists, merged via LDS at the end.
// ---------------------------------------------------------------------------
__launch_bounds__(64)
__global__ void dist_topk_kernel(const float* __restrict__ y,
                                 const float* __restrict__ sq,
                                 int* __restrict__ topkIdx) {
    __shared__ __attribute__((aligned(16))) float As[32][DIM];     // 32 KB
    __shared__ __attribute__((aligned(16))) float Bs[2][16][DIM];  // 2 x 16 KB
    __shared__ float Dts[2][16][16];                               // 2 KB
    __shared__ float mergeV[32][TOPK];
    __shared__ int   mergeI[32][TOPK];

    const int tid     = threadIdx.x;   // 0..63
    const int wave    = tid >> 5;
    const int lane    = tid & 31;
    const int half    = lane >> 4;
    const int l16     = lane & 15;
    const int batch   = blockIdx.y;
    const int rowbase = blockIdx.x * 32;

    const float* yb  = y + (size_t)batch * NPTS * DIM;
    const float* sqb = sq + batch * NPTS;

    const unsigned asBase = lds_offset_of(&As[0][0]);
    const unsigned bsBase[2] = { lds_offset_of(&Bs[0][0][0]),
                                 lds_offset_of(&Bs[1][0][0]) };

    // kick off async staging: the 32 A rows + B tile 0
    async_stage(yb + (size_t)rowbase * DIM, asBase, tid, 64, 32 * DIM * 4);
    async_stage(yb, bsBase[0], tid, 64, 16 * DIM * 4);

    float bestv[TOPK];
    int   besti[TOPK];
#pragma unroll
    for (int k = 0; k < TOPK; ++k) { bestv[k] = 3.4e38f; besti[k] = 0; }

    const int myrow = l16;          // row within this wave's 16
    const int chalf = half * 8;     // this lane scans tile cols [chalf, chalf+8)

    for (int jt = 0; jt < NPTS / 16; ++jt) {
        const int buf = jt & 1;
        async_wait0();                         // tile jt resident (own wave's loads)
        __syncthreads();                       // all waves' pieces visible
        if (jt + 1 < NPTS / 16)                // stream tile jt+1 under compute
            async_stage(yb + (size_t)(jt + 1) * 16 * DIM, bsBase[buf ^ 1],
                        tid, 64, 16 * DIM * 4);

        float sc = sqb[jt * 16 + l16];         // column norm for this lane's column

        v8f acc = {0.f, 0.f, 0.f, 0.f, 0.f, 0.f, 0.f, 0.f};
#pragma unroll 8
        for (int ks = 0; ks < DIM / 4; ++ks) {
            int kloc = ks * 4 + half * 2;
            v2f a = *(const v2f*)&As[wave * 16 + l16][kloc];
            v2f b = *(const v2f*)&Bs[buf][l16][kloc];
            acc = __builtin_amdgcn_wmma_f32_16x16x4_f32(
                false, a, false, b, (short)0, acc, false, false);
        }

        // D-tile -> LDS in (row, col) order; rank value = sq_j - 2*G
#pragma unroll
        for (int v = 0; v < 8; ++v)
            Dts[wave][half * 8 + v][l16] = sc - 2.0f * acc[v];
        __syncthreads();

        int colg = jt * 16 + chalf;
#pragma unroll
        for (int c = 0; c < 8; ++c)
            topk_insert(bestv, besti, Dts[wave][myrow][chalf + c], colg + c);
    }

    // merge the two half-lists per row
    __syncthreads();
    if (half == 1) {
#pragma unroll
        for (int k = 0; k < TOPK; ++k) {
            mergeV[wave * 16 + myrow][k] = bestv[k];
            mergeI[wave * 16 + myrow][k] = besti[k];
        }
    }
    __syncthreads();
    if (half == 0) {
#pragma unroll
        for (int k = 0; k < TOPK; ++k)
            topk_insert(bestv, besti, mergeV[wave * 16 + myrow][k],
                        mergeI[wave * 16 + myrow][k]);
        int row = batch * NPTS + rowbase + wave * 16 + myrow;
#pragma unroll
        for (int k = 0; k < TOPK; ++k)
            topkIdx[row * TOPK + k] = besti[k];
    }
}

// ---------------------------------------------------------------------------
// Kernel C: recompute logprobs = ||y_i - y_j||^2 * t exactly as the reference
// does, and emit the edge list. One wave per row; indices are read from the
// logprobs region and overwritten in place (read-before-write per wave).
// ---------------------------------------------------------------------------
__launch_bounds__(256)
__global__ void edges_logprobs_kernel(const float* __restrict__ y,
                                      const int* __restrict__ topkIdx,
                                      const float* __restrict__ temp,
                                      float* __restrict__ edges0,
                                      float* __restrict__ edges1,
                                      float* __restrict__ logprobs) {
    const int wv   = threadIdx.x >> 5;
    const int lane = threadIdx.x & 31;
    const int row  = blockIdx.x * 8 + wv;       // 0 .. BATCH*NPTS-1
    const int batch = row >> 12;

    float T = temp[0];
    T = fminf(fmaxf(T, -5.0f), 5.0f);
    const float t = __expf(T);

    const float* yi = y + (size_t)row * DIM;
    v4f a0 = *(const v4f*)(yi + lane * 8);
    v4f a1 = *(const v4f*)(yi + lane * 8 + 4);

#pragma unroll
    for (int k = 0; k < TOPK; ++k) {
        int j = topkIdx[row * TOPK + k];
        const float* yj = y + ((size_t)batch * NPTS + j) * DIM;
        v4f b0 = *(const v4f*)(yj + lane * 8);
        v4f b1 = *(const v4f*)(yj + lane * 8 + 4);
        float s = 0.0f;
        s += (a0.x - b0.x) * (a0.x - b0.x);
        s += (a0.y - b0.y) * (a0.y - b0.y);
        s += (a0.z - b0.z) * (a0.z - b0.z);
        s += (a0.w - b0.w) * (a0.w - b0.w);
        s += (a1.x - b1.x) * (a1.x - b1.x);
        s += (a1.y - b1.y) * (a1.y - b1.y);
        s += (a1.z - b1.z) * (a1.z - b1.z);
        s += (a1.w - b1.w) * (a1.w - b1.w);
#pragma unroll
        for (int off = 16; off > 0; off >>= 1) s += __shfl_xor(s, off, 32);
        if (lane == 0) {
            logprobs[row * TOPK + k] = s * t;                       // overwrites idx slot
            edges0[row * TOPK + k]   = (float)row;                  // i + batch*n
            edges1[row * TOPK + k]   = (float)(batch * NPTS + j);   // idx + batch*n
        }
    }
}

// ---------------------------------------------------------------------------
extern "C" void kernel_launch(void* const* d_in, const int* in_sizes, int n_in,
                              void* d_out, int out_size, void* d_ws, size_t ws_size,
                              hipStream_t stream) {
    (void)in_sizes; (void)n_in; (void)out_size; (void)d_ws; (void)ws_size;

    const float* x    = (const float*)d_in[0];
    const float* W    = (const float*)d_in[1];
    const float* temp = (const float*)d_in[2];

    float* out = (float*)d_out;
    const size_t yCount    = (size_t)BATCH * NPTS * DIM;          // 8,388,608
    const size_t edgeCount = (size_t)BATCH * NPTS * TOPK;         // 327,680
    float* y      = out;
    float* edges0 = out + yCount;
    float* edges1 = edges0 + edgeCount;
    float* logp   = edges1 + edgeCount;

    // Scratch aliases inside d_out (rewritten before final use each call):
    float* sq      = edges0;        // 32,768 floats << edges region
    int*   topkIdx = (int*)logp;    // exactly the logprobs region

    dim3 gA(NPTS / 16, BATCH);
    embed_gemm_kernel<<<gA, 128, 0, stream>>>(x, W, y, sq);

    dim3 gB(NPTS / 32, BATCH);
    dist_topk_kernel<<<gB, 64, 0, stream>>>(y, sq, topkIdx);

    edges_logprobs_kernel<<<(BATCH * NPTS) / 8, 256, 0, stream>>>(
        y, topkIdx, temp, edges0, edges1, logp);
}